// GCN_deeper_7687991460117
// MI455X (gfx1250) — compile-verified
//
#include <hip/hip_runtime.h>
#include <hip/hip_bf16.h>

#define NUM_NODES  100000
#define NUM_EDGES  3200000
#define NUM_GRAPHS 200
#define F_IN       128
#define HID        64
#define BN_EPS     1e-5f

typedef __attribute__((ext_vector_type(2))) float v2f;
typedef __attribute__((ext_vector_type(8))) float v8f;

// ---------------------------------------------------------------------------
// 1) init: deg = 1.0 (self loop), zero pooling buffers
// ---------------------------------------------------------------------------
__global__ void init_kernel(float* __restrict__ deg,
                            float* __restrict__ sums,
                            float* __restrict__ cnts) {
    int i = blockIdx.x * blockDim.x + threadIdx.x;
    if (i < NUM_NODES)        deg[i]  = 1.0f;
    if (i < NUM_GRAPHS * HID) sums[i] = 0.0f;
    if (i < NUM_GRAPHS)       cnts[i] = 0.0f;
}

// ---------------------------------------------------------------------------
// 2) degree over dst
// ---------------------------------------------------------------------------
__global__ void deg_kernel(const int* __restrict__ ei, float* __restrict__ deg) {
    int e = blockIdx.x * blockDim.x + threadIdx.x;
    if (e < NUM_EDGES) {
        int dst = ei[NUM_EDGES + e];
        atomicAdd(&deg[dst], 1.0f);
    }
}

// ---------------------------------------------------------------------------
// 3) dinv = rsqrt(deg), in place
// ---------------------------------------------------------------------------
__global__ void dinv_kernel(float* __restrict__ deg) {
    int i = blockIdx.x * blockDim.x + threadIdx.x;
    if (i < NUM_NODES) deg[i] = rsqrtf(deg[i]);
}

// ---------------------------------------------------------------------------
// 4/8) H[N,64] = X[N,K] @ W[64,K]^T with V_WMMA_F32_16X16X4_F32.
// One wave produces a 16(M) x 64(N) slab: 4 accumulators share each A fetch.
// Tile-count guard is wave-uniform -> EXEC all-ones at every WMMA.
// ---------------------------------------------------------------------------
template <int K>
__global__ __launch_bounds__(256) void gemm_wmma(const float* __restrict__ X,
                                                 const float* __restrict__ W,
                                                 float* __restrict__ H) {
    const int wave = (blockIdx.x * blockDim.x + threadIdx.x) >> 5; // global wave id
    if (wave >= NUM_NODES / 16) return;                            // uniform per wave

    const int lane = threadIdx.x & 31;
    const int half = lane >> 4;   // 0: lanes 0-15, 1: lanes 16-31
    const int r    = lane & 15;
    const int m0   = wave * 16;

    v8f c0 = {}, c1 = {}, c2 = {}, c3 = {};

    const float* xrow = X + (size_t)(m0 + r) * K;  // A: row M = r (per half-lane group)
    const float* w0   = W + (size_t)(r)      * K;  // B: column N = r of each N-tile
    const float* w1   = W + (size_t)(16 + r) * K;
    const float* w2   = W + (size_t)(32 + r) * K;
    const float* w3   = W + (size_t)(48 + r) * K;

#pragma unroll
    for (int k = 0; k < K; k += 4) {
        const int ko = k + 2 * half;  // lanes 0-15: K=k,k+1 ; lanes 16-31: K=k+2,k+3
        v2f a  = *(const v2f*)(xrow + ko);
        v2f b0 = *(const v2f*)(w0 + ko);
        v2f b1 = *(const v2f*)(w1 + ko);
        v2f b2 = *(const v2f*)(w2 + ko);
        v2f b3 = *(const v2f*)(w3 + ko);
        c0 = __builtin_amdgcn_wmma_f32_16x16x4_f32(false, a, false, b0, (short)0, c0, false, false);
        c1 = __builtin_amdgcn_wmma_f32_16x16x4_f32(false, a, false, b1, (short)0, c1, false, false);
        c2 = __builtin_amdgcn_wmma_f32_16x16x4_f32(false, a, false, b2, (short)0, c2, false, false);
        c3 = __builtin_amdgcn_wmma_f32_16x16x4_f32(false, a, false, b3, (short)0, c3, false, false);
    }

    // C/D layout: VGPR j -> M = m0 + j + 8*half, N = r (within each 16-wide N tile)
    float* hrow = H + (size_t)m0 * HID;
#pragma unroll
    for (int j = 0; j < 8; ++j) {
        const size_t row = (size_t)(j + 8 * half) * HID;
        hrow[row + 0  + r] = c0[j];
        hrow[row + 16 + r] = c1[j];
        hrow[row + 32 + r] = c2[j];
        hrow[row + 48 + r] = c3[j];
    }
}

// ---------------------------------------------------------------------------
// 5/9) agg = h * dinv^2 + bias   (self-loop term + bias folded in)
// ---------------------------------------------------------------------------
__global__ void self_bias_kernel(const float* __restrict__ h,
                                 const float* __restrict__ dinv,
                                 const float* __restrict__ bias,
                                 float* __restrict__ agg) {
    int i = blockIdx.x * blockDim.x + threadIdx.x;
    if (i < NUM_NODES * HID) {
        int n = i >> 6, f = i & 63;
        float d = dinv[n];
        agg[i] = h[i] * d * d + bias[f];
    }
}

// ---------------------------------------------------------------------------
// 6/10) agg[dst] += h[src] * dinv[src]*dinv[dst]  (64 lanes per edge)
// Both h (26 MB) and agg (26 MB) are L2-resident on MI455X (192 MB L2).
// ---------------------------------------------------------------------------
__global__ void edge_scatter_kernel(const int* __restrict__ ei,
                                    const float* __restrict__ h,
                                    const float* __restrict__ dinv,
                                    float* __restrict__ agg) {
    int idx = blockIdx.x * blockDim.x + threadIdx.x; // < 204.8M, fits int32
    int e = idx >> 6, f = idx & 63;
    if (e < NUM_EDGES) {
        int src = ei[e];
        int dst = ei[NUM_EDGES + e];
        float w = dinv[src] * dinv[dst];
        atomicAdd(&agg[(size_t)dst * HID + f], h[(size_t)src * HID + f] * w);
    }
}

// ---------------------------------------------------------------------------
// 7/11) BN (eval) + ReLU, in place
// ---------------------------------------------------------------------------
__global__ void bn_relu_kernel(float* __restrict__ agg,
                               const float* __restrict__ gamma,
                               const float* __restrict__ beta,
                               const float* __restrict__ mean,
                               const float* __restrict__ var) {
    int i = blockIdx.x * blockDim.x + threadIdx.x;
    if (i < NUM_NODES * HID) {
        int f = i & 63;
        float s = gamma[f] * rsqrtf(var[f] + BN_EPS);
        float v = (agg[i] - mean[f]) * s + beta[f];
        agg[i] = fmaxf(v, 0.0f);
    }
}

// ---------------------------------------------------------------------------
// 12) pooled sums + counts (batch is sorted; 12.8K bins, L2-resident)
// ---------------------------------------------------------------------------
__global__ void pool_scatter_kernel(const float* __restrict__ h,
                                    const int* __restrict__ batch,
                                    float* __restrict__ sums,
                                    float* __restrict__ cnts) {
    int i = blockIdx.x * blockDim.x + threadIdx.x;
    if (i < NUM_NODES * HID) {
        int n = i >> 6, f = i & 63;
        int g = batch[n];
        atomicAdd(&sums[g * HID + f], h[i]);
        if (f == 0) atomicAdd(&cnts[g], 1.0f);
    }
}

// ---------------------------------------------------------------------------
// 13) out[g] = (sum_f pooled[g][f]*Wl[f]) + bl
// ---------------------------------------------------------------------------
__global__ void final_kernel(const float* __restrict__ sums,
                             const float* __restrict__ cnts,
                             const float* __restrict__ Wl,
                             const float* __restrict__ bl,
                             float* __restrict__ out) {
    int g = blockIdx.x * blockDim.x + threadIdx.x;
    if (g < NUM_GRAPHS) {
        float inv = 1.0f / fmaxf(cnts[g], 1.0f);
        float acc = 0.0f;
        for (int f = 0; f < HID; ++f) acc += sums[g * HID + f] * Wl[f];
        out[g] = acc * inv + bl[0];
    }
}

extern "C" void kernel_launch(void* const* d_in, const int* in_sizes, int n_in,
                              void* d_out, int out_size, void* d_ws, size_t ws_size,
                              hipStream_t stream) {
    const float* x      = (const float*)d_in[0];
    const int*   ei     = (const int*)  d_in[1];
    const int*   batch  = (const int*)  d_in[2];
    const float* W1     = (const float*)d_in[3];
    const float* b1     = (const float*)d_in[4];
    const float* gamma1 = (const float*)d_in[5];
    const float* beta1  = (const float*)d_in[6];
    const float* mean1  = (const float*)d_in[7];
    const float* var1   = (const float*)d_in[8];
    const float* W2     = (const float*)d_in[9];
    const float* b2     = (const float*)d_in[10];
    const float* gamma2 = (const float*)d_in[11];
    const float* beta2  = (const float*)d_in[12];
    const float* mean2  = (const float*)d_in[13];
    const float* var2   = (const float*)d_in[14];
    const float* Wl     = (const float*)d_in[15];
    const float* bl     = (const float*)d_in[16];
    float* out = (float*)d_out;

    // workspace layout (~55.1 MB)
    char* ws = (char*)d_ws;
    const size_t MB = 1u << 20;
    float* dinv = (float*)(ws + 0 * MB);           // 0.4 MB (deg, then dinv in place)
    float* A    = (float*)(ws + 1 * MB);           // 25.6 MB  h (pre-agg)
    float* B    = (float*)(ws + 28 * MB);          // 25.6 MB  agg / activations
    float* sums = (float*)(ws + 55 * MB);          // 50 KB
    float* cnts = (float*)(ws + 55 * MB + 65536);  // 0.8 KB

    const int T = 256;
    const int gN   = (NUM_NODES + T - 1) / T;            // node-wise
    const int gNF  = (NUM_NODES * HID + T - 1) / T;      // node*feature
    const int gE   = (NUM_EDGES + T - 1) / T;            // edge-wise
    const int gEF  = (NUM_EDGES * HID + T - 1) / T;      // edge*feature (800000)
    const int gGEMM = (NUM_NODES / 16 + 7) / 8;          // 8 waves/block

    // degrees
    init_kernel<<<gN, T, 0, stream>>>(dinv, sums, cnts);
    deg_kernel<<<gE, T, 0, stream>>>(ei, dinv);
    dinv_kernel<<<gN, T, 0, stream>>>(dinv);

    // ---- layer 1 ----
    gemm_wmma<F_IN><<<gGEMM, T, 0, stream>>>(x, W1, A);
    self_bias_kernel<<<gNF, T, 0, stream>>>(A, dinv, b1, B);
    edge_scatter_kernel<<<gEF, T, 0, stream>>>(ei, A, dinv, B);
    bn_relu_kernel<<<gNF, T, 0, stream>>>(B, gamma1, beta1, mean1, var1);

    // ---- layer 2 ----
    gemm_wmma<HID><<<gGEMM, T, 0, stream>>>(B, W2, A);
    self_bias_kernel<<<gNF, T, 0, stream>>>(A, dinv, b2, B);
    edge_scatter_kernel<<<gEF, T, 0, stream>>>(ei, A, dinv, B);
    bn_relu_kernel<<<gNF, T, 0, stream>>>(B, gamma2, beta2, mean2, var2);

    // ---- pool + linear head ----
    pool_scatter_kernel<<<gNF, T, 0, stream>>>(B, batch, sums, cnts);
    final_kernel<<<1, T, 0, stream>>>(sums, cnts, Wl, bl, out);
}